// ContrastiveHessianCalculator_42365557408120
// MI455X (gfx1250) — compile-verified
//
#include <hip/hip_runtime.h>
#include <math.h>

// Problem constants (match reference)
#define D_ 256
#define H_ 512
#define O_ 64
#define P_ 1024
#define NROWS (4 * P_)   // 4096 gathered rows (ap, p, an, n)
#define KU (3 * 2 * P_)  // 6144 K-rows for the second GEMM

typedef __attribute__((ext_vector_type(2))) float v2f;
typedef __attribute__((ext_vector_type(8))) float v8f;

// ---------------------------------------------------------------------------
// K0: c[i] = sum_o W2[o][i]^2 ; also zero the b2 block of the output.
// ---------------------------------------------------------------------------
__global__ void k0_precompute_c(const float* __restrict__ W2,
                                float* __restrict__ c,
                                float* __restrict__ out) {
    int i = threadIdx.x + blockIdx.x * blockDim.x;
    if (i < H_) {
        float s = 0.f;
        for (int o = 0; o < O_; ++o) {
            float w = W2[(size_t)o * H_ + i];
            s += w * w;
        }
        c[i] = s;
    }
    if (i < O_) {
        out[(size_t)H_ * D_ + H_ + (size_t)O_ * H_ + i] = 0.f;  // b2 diag = 0
    }
}

// ---------------------------------------------------------------------------
// K1: forward pass with f32 WMMA, register-blocked 1x4 (16x64 strip / wave).
//   Z[4096x512] = Xgather[4096x256] * W1^T[256x512] + b1 ; Hbuf = tanh(Z)
// V_WMMA_F32_16X16X4_F32 layouts (ISA 7.12.2):
//   A 16x4: lanes 0-15 -> M=lane, K={k0,k0+1}; lanes 16-31 -> M=lane-16,
//           K={k0+2,k0+3}.
//   B 4x16: VGPR0/1 = rows {k0,k0+1} (lanes 0-15), {k0+2,k0+3} (lanes 16-31),
//           N = lane&15.
//   C/D:    VGPR r -> M = r (lanes 0-15) / r+8 (lanes 16-31), N = lane&15.
// The A operand (per-lane gathered x row) is loaded once per K-step and
// reused by 4 WMMAs (one per 16-wide N sub-tile).
// ---------------------------------------------------------------------------
__global__ void k1_forward_wmma(const float* __restrict__ x,
                                const float* __restrict__ W1,
                                const float* __restrict__ b1,
                                const int* __restrict__ ap,
                                const int* __restrict__ p,
                                const int* __restrict__ an,
                                const int* __restrict__ n,
                                float* __restrict__ Hbuf) {
    const int lane = threadIdx.x & 31;
    const int half = lane >> 4;
    const int l16  = lane & 15;
    const int m0 = blockIdx.x * 16;   // gathered-row tile
    const int n0 = blockIdx.y * 64;   // 64-wide H strip (4 sub-tiles)

    // Per-lane gathered A row: row m0+l16 of the virtual [4096 x D] matrix.
    int m = m0 + l16;
    int s = m >> 10;        // which index array
    int t = m & (P_ - 1);
    int xr = (s == 0) ? ap[t] : (s == 1) ? p[t] : (s == 2) ? an[t] : n[t];
    const float* __restrict__ xrow = x + (size_t)xr * D_;
    // B[k][n] = W1[n][k] -> column n of W1^T is row n of W1.
    const float* __restrict__ wrow = W1 + (size_t)(n0 + l16) * D_;

    v8f acc[4] = {{}, {}, {}, {}};
    for (int k0 = 0; k0 < D_; k0 += 4) {
        const int kk = k0 + half * 2;
        v2f a;
        a.x = xrow[kk];
        a.y = xrow[kk + 1];
#pragma unroll
        for (int tt = 0; tt < 4; ++tt) {
            v2f b;
            b.x = wrow[(size_t)(16 * tt) * D_ + kk];
            b.y = wrow[(size_t)(16 * tt) * D_ + kk + 1];
            acc[tt] = __builtin_amdgcn_wmma_f32_16x16x4_f32(
                false, a, false, b, (short)0, acc[tt], false, false);
        }
    }

#pragma unroll
    for (int tt = 0; tt < 4; ++tt) {
        const int nn = n0 + 16 * tt + l16;
        const float bias = b1[nn];
#pragma unroll
        for (int r = 0; r < 8; ++r) {
            int mm = m0 + r + half * 8;
            Hbuf[(size_t)mm * H_ + nn] = tanhf(acc[tt][r] + bias);
        }
    }
}

// ---------------------------------------------------------------------------
// K2: build U[6144x512] and V[6144x256] for the second GEMM.
//   pair q (0..2047): sign=+1 for q<P_ (pos), -1 otherwise (neg).
//   U rows 3q..3q+2 = sign*S11, -2*sign*S12, sign*S22   (S = d_a d_b c)
//   V rows 3q..3q+2 = xa^2, xa*xb, xb^2
// ---------------------------------------------------------------------------
__global__ void k2_build_uv(const float* __restrict__ x,
                            const float* __restrict__ Hbuf,
                            const float* __restrict__ c,
                            const int* __restrict__ ap,
                            const int* __restrict__ p,
                            const int* __restrict__ an,
                            const int* __restrict__ n,
                            float* __restrict__ U,
                            float* __restrict__ V) {
    const int q   = blockIdx.x;      // 0..2047
    const int tid = threadIdx.x;     // 0..255
    const bool pos = (q < P_);
    const int qq = pos ? q : q - P_;
    const float sign = pos ? 1.f : -1.f;
    const int rowA = pos ? qq : (2 * P_ + qq);
    const int rowB = pos ? (P_ + qq) : (3 * P_ + qq);

    const float* __restrict__ hA = Hbuf + (size_t)rowA * H_;
    const float* __restrict__ hB = Hbuf + (size_t)rowB * H_;
    float* __restrict__ U0 = U + (size_t)(3 * q) * H_;

    for (int i = tid; i < H_; i += 256) {
        float ha = hA[i], hb = hB[i];
        float da = 1.f - ha * ha;
        float db = 1.f - hb * hb;
        float ci = c[i];
        U0[i]          = sign * da * da * ci;
        U0[H_ + i]     = -2.f * sign * da * db * ci;
        U0[2 * H_ + i] = sign * db * db * ci;
    }

    const int ia = pos ? ap[qq] : an[qq];
    const int ib = pos ? p[qq]  : n[qq];
    const float* __restrict__ xa = x + (size_t)ia * D_;
    const float* __restrict__ xb = x + (size_t)ib * D_;
    float* __restrict__ V0 = V + (size_t)(3 * q) * D_;

    for (int j = tid; j < D_; j += 256) {
        float va = xa[j], vb = xb[j];
        V0[j]          = va * va;
        V0[D_ + j]     = va * vb;
        V0[2 * D_ + j] = vb * vb;
    }
}

// ---------------------------------------------------------------------------
// K3: W1d[512x256] = U^T * V via f32 WMMA, K = 6144, 2x4 register blocking
// (32x64 output per wave). A[m][k] = U[k][i0+m] (uncoalesced column access,
// loaded once per K-step per i-sub-tile, reused by 4 WMMAs); B[k][j] = V[k][j].
// Result written directly into the W1-diag block of out.
// ---------------------------------------------------------------------------
__global__ void k3_gemm2_wmma(const float* __restrict__ U,
                              const float* __restrict__ V,
                              float* __restrict__ outW1) {
    const int lane = threadIdx.x & 31;
    const int half = lane >> 4;
    const int l16  = lane & 15;
    const int i0 = blockIdx.x * 32;   // 32-row H strip (2 sub-tiles)
    const int j0 = blockIdx.y * 64;   // 64-col D strip (4 sub-tiles)

    v8f acc[2][4] = {{{}, {}, {}, {}}, {{}, {}, {}, {}}};
    for (int k0 = 0; k0 < KU; k0 += 4) {
        const int kk = k0 + half * 2;
        v2f a[2];
#pragma unroll
        for (int ii = 0; ii < 2; ++ii) {
            a[ii].x = U[(size_t)(kk)     * H_ + i0 + 16 * ii + l16];
            a[ii].y = U[(size_t)(kk + 1) * H_ + i0 + 16 * ii + l16];
        }
#pragma unroll
        for (int tt = 0; tt < 4; ++tt) {
            v2f b;
            b.x = V[(size_t)(kk)     * D_ + j0 + 16 * tt + l16];
            b.y = V[(size_t)(kk + 1) * D_ + j0 + 16 * tt + l16];
#pragma unroll
            for (int ii = 0; ii < 2; ++ii) {
                acc[ii][tt] = __builtin_amdgcn_wmma_f32_16x16x4_f32(
                    false, a[ii], false, b, (short)0, acc[ii][tt], false, false);
            }
        }
    }

#pragma unroll
    for (int ii = 0; ii < 2; ++ii) {
#pragma unroll
        for (int tt = 0; tt < 4; ++tt) {
#pragma unroll
            for (int r = 0; r < 8; ++r) {
                int i = i0 + 16 * ii + r + half * 8;
                outW1[(size_t)i * D_ + j0 + 16 * tt + l16] = acc[ii][tt][r];
            }
        }
    }
}

// ---------------------------------------------------------------------------
// K4: tail reductions.
//   b1d[i] = column sum of U (sign and -2 factor already folded in).
//   g[i]   = sum_q sign*(ha-hb)^2 ; W2d[o][i] = g[i] for every o.
// ---------------------------------------------------------------------------
__global__ void k4_tail(const float* __restrict__ U,
                        const float* __restrict__ Hbuf,
                        float* __restrict__ out) {
    int i = threadIdx.x + blockIdx.x * blockDim.x;  // 0..511
    if (i >= H_) return;

    float b1d = 0.f;
    for (int k = 0; k < KU; ++k) b1d += U[(size_t)k * H_ + i];
    out[(size_t)H_ * D_ + i] = b1d;

    float g = 0.f;
    for (int q = 0; q < P_; ++q) {
        float dp = Hbuf[(size_t)q * H_ + i]            - Hbuf[(size_t)(P_ + q) * H_ + i];
        float dn = Hbuf[(size_t)(2 * P_ + q) * H_ + i] - Hbuf[(size_t)(3 * P_ + q) * H_ + i];
        g += dp * dp - dn * dn;
    }
    const size_t w2off = (size_t)H_ * D_ + H_;
    for (int o = 0; o < O_; ++o) out[w2off + (size_t)o * H_ + i] = g;
}

// ---------------------------------------------------------------------------
extern "C" void kernel_launch(void* const* d_in, const int* in_sizes, int n_in,
                              void* d_out, int out_size, void* d_ws, size_t ws_size,
                              hipStream_t stream) {
    const float* x  = (const float*)d_in[0];
    const float* W1 = (const float*)d_in[1];
    const float* b1 = (const float*)d_in[2];
    const float* W2 = (const float*)d_in[3];
    // d_in[4] = b2 (unused: its Hessian diag is identically zero)
    const int* ap = (const int*)d_in[5];
    const int* p  = (const int*)d_in[6];
    const int* an = (const int*)d_in[7];
    const int* nn = (const int*)d_in[8];

    float* out = (float*)d_out;
    float* ws  = (float*)d_ws;

    // Workspace layout (floats): c[512] | Hbuf[4096*512] | U[6144*512] | V[6144*256]
    float* c    = ws;
    float* Hbuf = c + H_;
    float* U    = Hbuf + (size_t)NROWS * H_;
    float* V    = U + (size_t)KU * H_;

    k0_precompute_c<<<2, 256, 0, stream>>>(W2, c, out);
    k1_forward_wmma<<<dim3(NROWS / 16, H_ / 64), 32, 0, stream>>>(
        x, W1, b1, ap, p, an, nn, Hbuf);
    k2_build_uv<<<2 * P_, 256, 0, stream>>>(x, Hbuf, c, ap, p, an, nn, U, V);
    k3_gemm2_wmma<<<dim3(H_ / 32, D_ / 64), 32, 0, stream>>>(U, V, out);
    k4_tail<<<2, 256, 0, stream>>>(U, Hbuf, out);
}